// GCNlayer_76776835384057
// MI455X (gfx1250) — compile-verified
//
#include <hip/hip_runtime.h>
#include <hip/hip_bf16.h>

typedef __attribute__((ext_vector_type(2))) float v2f;
typedef __attribute__((ext_vector_type(8))) float v8f;

#define N_NODES 100000
#define N_EDGES 1600000
#define N_FEAT  75
#define PAD_K   76          // 75 rounded up to multiple of 4
#define HID     64
#define NUM_GRAPHS 4096
#define BN_EPS  1e-5f

// ---------------------------------------------------------------------------
// Utility: zero a float buffer
// ---------------------------------------------------------------------------
__global__ void zero_f32(float* __restrict__ p, long n) {
    long t = (long)blockIdx.x * blockDim.x + threadIdx.x;
    if (t < n) p[t] = 0.0f;
}

// ---------------------------------------------------------------------------
// Pad x [N x 75] -> xp [N x 76] (zero tail) so GEMM A-loads are unguarded
// aligned 64-bit loads.
// ---------------------------------------------------------------------------
__global__ void pad_x(const float* __restrict__ x, float* __restrict__ xp) {
    long t = (long)blockIdx.x * blockDim.x + threadIdx.x;
    if (t >= (long)N_NODES * PAD_K) return;
    int n = (int)(t / PAD_K);
    int k = (int)(t % PAD_K);
    xp[t] = (k < N_FEAT) ? x[(long)n * N_FEAT + k] : 0.0f;
}

// Pad W1 [75 x 64] -> [76 x 64] (zero last row)
__global__ void pad_w1(const float* __restrict__ w, float* __restrict__ wp) {
    int t = blockIdx.x * blockDim.x + threadIdx.x;
    if (t >= PAD_K * HID) return;
    int k = t >> 6;
    wp[t] = (k < N_FEAT) ? w[t] : 0.0f;
}

// ---------------------------------------------------------------------------
// Degree: deg = 1 (self loop) + in-degree ;  dinv = rsqrt(deg)
// ---------------------------------------------------------------------------
__global__ void init_deg(float* __restrict__ deg, int n) {
    int t = blockIdx.x * blockDim.x + threadIdx.x;
    if (t < n) deg[t] = 1.0f;
}

__global__ void accum_deg(const int* __restrict__ dst, float* __restrict__ deg, int ne) {
    int t = blockIdx.x * blockDim.x + threadIdx.x;
    if (t < ne) unsafeAtomicAdd(&deg[dst[t]], 1.0f);
}

__global__ void compute_dinv(const float* __restrict__ deg, float* __restrict__ dinv, int n) {
    int t = blockIdx.x * blockDim.x + threadIdx.x;
    if (t < n) dinv[t] = rsqrtf(deg[t]);
}

// ---------------------------------------------------------------------------
// GEMM: H[N x 64] = X[N x DIN] @ W[DIN x 64], fp32 WMMA 16x16x4.
// DIN is a compile-time multiple of 4 (64 or 76) == row stride of X.
// One wave = one 16x16 tile of H. Block = 128 threads (4 waves = 4 col tiles),
// blockIdx.x = 16-row strip (N = 6250 * 16 exactly -> no row guards).
// W is staged in LDS as interleaved (W[k][c], W[k+1][c]) pairs so each lane's
// B fragment is a single ds_load_b64; A fragment is a single global_load_b64.
// A layout (16x4 f32): lanes 0-15 M=0..15 {K=0,1}, lanes 16-31 {K=2,3}.
// B layout (4x16 f32): lanes 0-15 N=0..15 {K=0,1}, lanes 16-31 {K=2,3}.
// C layout (16x16 f32): VGPR i -> M=i (lanes 0-15) / M=i+8 (lanes 16-31).
// ---------------------------------------------------------------------------
template <int DIN>
__global__ __launch_bounds__(128) void gemm_wmma_f32(
        const float* __restrict__ X, const float* __restrict__ W,
        float* __restrict__ H) {
    __shared__ float2 wlds[(DIN / 2) * HID];

    const int tid  = threadIdx.x;
    const int lane = tid & 31;
    const int wid  = tid >> 5;              // column tile 0..3
    const int row0 = blockIdx.x * 16;
    const int col0 = wid * 16;
    const int lh   = lane & 15;
    const int khalf = (lane >> 4) << 1;     // 0 for lanes 0-15, 2 for 16-31

    // cooperative stage of W into LDS (interleaved k-pairs)
    for (int i = tid; i < DIN * HID; i += 128) {
        const int k = i >> 6;
        const int c = i & 63;
        const float v = W[i];
        if (k & 1) wlds[(k >> 1) * HID + c].y = v;
        else       wlds[(k >> 1) * HID + c].x = v;
    }
    __syncthreads();

    const float* xrow = X + (long)(row0 + lh) * DIN;
    const int col = col0 + lh;

    v8f c = {};
    #pragma unroll
    for (int k0 = 0; k0 < DIN; k0 += 4) {
        const int kk = k0 + khalf;                    // even
        const v2f a = *(const v2f*)(xrow + kk);       // global_load_b64
        const float2 bp = wlds[(kk >> 1) * HID + col]; // ds_load_b64
        v2f b;
        b.x = bp.x;
        b.y = bp.y;
        // D = A*B + C   (v_wmma_f32_16x16x4_f32)
        c = __builtin_amdgcn_wmma_f32_16x16x4_f32(
                false, a, false, b, (short)0, c, false, false);
    }

    const int rbase = row0 + ((lane >> 4) << 3);      // +8 rows for upper half
    #pragma unroll
    for (int i = 0; i < 8; ++i) {
        H[(long)(rbase + i) * HID + col] = c[i];
    }
}

// ---------------------------------------------------------------------------
// Edge scatter: agg[dst] += h[src] * (dinv[src]*dinv[dst])
// One thread handles one edge x 4 features (aligned float4 gather).
// ---------------------------------------------------------------------------
__global__ __launch_bounds__(256) void edge_scatter(
        const float* __restrict__ lin, const int* __restrict__ src,
        const int* __restrict__ dst, const float* __restrict__ dinv,
        float* __restrict__ agg, int ne) {
    long t = (long)blockIdx.x * blockDim.x + threadIdx.x;
    int e  = (int)(t >> 4);
    if (e >= ne) return;
    int fo = (int)(t & 15) << 2;

    // prefetch index streams ahead (global_prefetch_b8, speculative-safe)
    __builtin_prefetch(&src[e + 512], 0, 1);
    __builtin_prefetch(&dst[e + 512], 0, 1);

    const int s = src[e];
    const int d = dst[e];
    const float en = dinv[s] * dinv[d];
    const float4 hv = *(const float4*)(lin + (long)s * HID + fo);
    float* ap = agg + (long)d * HID + fo;
    unsafeAtomicAdd(ap + 0, hv.x * en);
    unsafeAtomicAdd(ap + 1, hv.y * en);
    unsafeAtomicAdd(ap + 2, hv.z * en);
    unsafeAtomicAdd(ap + 3, hv.w * en);
}

// ---------------------------------------------------------------------------
// Combine: act = relu(agg + lin*self_norm + b); accumulate BN stats
// (column sum & sumsq) via LDS ds_add_f32, one global atomic per column/block.
// Block = 256 threads = 4 nodes x 64 features.
// ---------------------------------------------------------------------------
__global__ __launch_bounds__(256) void combine_relu_stats(
        const float* __restrict__ lin, const float* __restrict__ agg,
        const float* __restrict__ dinv, const float* __restrict__ bias,
        float* __restrict__ act, float* __restrict__ stats, int n) {
    __shared__ float ssum[HID];
    __shared__ float ssq[HID];
    const int t = threadIdx.x;
    if (t < HID) { ssum[t] = 0.0f; ssq[t] = 0.0f; }
    __syncthreads();

    const int node = blockIdx.x * 4 + (t >> 6);
    const int f    = t & 63;
    float y = 0.0f;
    if (node < n) {
        float sn = dinv[node];
        sn *= sn;
        const long idx = (long)node * HID + f;
        y = agg[idx] + lin[idx] * sn + bias[f];
        y = fmaxf(y, 0.0f);
        act[idx] = y;
    }
    unsafeAtomicAdd(&ssum[f], y);
    unsafeAtomicAdd(&ssq[f], y * y);
    __syncthreads();
    if (t < HID) {
        unsafeAtomicAdd(&stats[t],       ssum[t]);
        unsafeAtomicAdd(&stats[HID + t], ssq[t]);
    }
}

// ---------------------------------------------------------------------------
// BN apply (training mode, biased variance): in-place on act
// ---------------------------------------------------------------------------
__global__ __launch_bounds__(256) void bn_apply(
        float* __restrict__ act, const float* __restrict__ stats,
        const float* __restrict__ g, const float* __restrict__ be, int n) {
    long t = (long)blockIdx.x * blockDim.x + threadIdx.x;
    if (t >= (long)n * HID) return;
    const int f = (int)(t & 63);
    const float inv_n = 1.0f / (float)n;
    const float m = stats[f] * inv_n;
    const float v = stats[HID + f] * inv_n - m * m;
    act[t] = (act[t] - m) * rsqrtf(v + BN_EPS) * g[f] + be[f];
}

// ---------------------------------------------------------------------------
// Global add pool: out[batch[n]] += act[n]
// ---------------------------------------------------------------------------
__global__ __launch_bounds__(256) void pool_scatter(
        const float* __restrict__ act, const int* __restrict__ batch,
        float* __restrict__ out, int n) {
    long t = (long)blockIdx.x * blockDim.x + threadIdx.x;
    if (t >= (long)n * HID) return;
    const int node = (int)(t >> 6);
    const int f    = (int)(t & 63);
    unsafeAtomicAdd(&out[(long)batch[node] * HID + f], act[t]);
}

// ---------------------------------------------------------------------------
// Host-side orchestration
// ---------------------------------------------------------------------------
extern "C" void kernel_launch(void* const* d_in, const int* in_sizes, int n_in,
                              void* d_out, int out_size, void* d_ws, size_t ws_size,
                              hipStream_t stream) {
    (void)in_sizes; (void)n_in; (void)out_size; (void)ws_size;

    const float* x     = (const float*)d_in[0];
    const int*   src   = (const int*)  d_in[1];
    const int*   dst   = (const int*)  d_in[2];
    const int*   batch = (const int*)  d_in[3];
    const float* Wl[4]  = { (const float*)d_in[4],  (const float*)d_in[8],
                            (const float*)d_in[12], (const float*)d_in[16] };
    const float* bl[4]  = { (const float*)d_in[5],  (const float*)d_in[9],
                            (const float*)d_in[13], (const float*)d_in[17] };
    const float* gl[4]  = { (const float*)d_in[6],  (const float*)d_in[10],
                            (const float*)d_in[14], (const float*)d_in[18] };
    const float* bel[4] = { (const float*)d_in[7],  (const float*)d_in[11],
                            (const float*)d_in[15], (const float*)d_in[19] };
    float* out = (float*)d_out;

    // workspace layout (floats)
    float* wsf   = (float*)d_ws;
    float* deg   = wsf;                                // N
    float* dinv  = deg  + N_NODES;                     // N
    float* lin   = dinv + N_NODES;                     // N*64
    float* agg   = lin  + (long)N_NODES * HID;         // N*64
    float* act   = agg  + (long)N_NODES * HID;         // N*64
    float* stats = act  + (long)N_NODES * HID;         // 128
    float* xp    = stats + 2 * HID;                    // N*76
    float* w1p   = xp   + (long)N_NODES * PAD_K;       // 76*64

    const int B = 256;
    const long nh   = (long)N_NODES * HID;
    const int  gNH  = (int)((nh + B - 1) / B);
    const int  gN   = (N_NODES + B - 1) / B;
    const int  gE   = (N_EDGES + B - 1) / B;
    const int  gE16 = (int)(((long)N_EDGES * 16 + B - 1) / B);

    // one-time prep: padded x / W1, degree + dinv
    pad_x<<<(int)(((long)N_NODES * PAD_K + B - 1) / B), B, 0, stream>>>(x, xp);
    pad_w1<<<(PAD_K * HID + B - 1) / B, B, 0, stream>>>(Wl[0], w1p);
    init_deg<<<gN, B, 0, stream>>>(deg, N_NODES);
    accum_deg<<<gE, B, 0, stream>>>(dst, deg, N_EDGES);
    compute_dinv<<<gN, B, 0, stream>>>(deg, dinv, N_NODES);

    // 4 GCN layers
    for (int l = 0; l < 4; ++l) {
        if (l == 0) {
            gemm_wmma_f32<PAD_K><<<N_NODES / 16, 128, 0, stream>>>(xp, w1p, lin);
        } else {
            gemm_wmma_f32<HID><<<N_NODES / 16, 128, 0, stream>>>(act, Wl[l], lin);
        }
        zero_f32<<<gNH, B, 0, stream>>>(agg, nh);
        zero_f32<<<1, B, 0, stream>>>(stats, 2 * HID);
        edge_scatter<<<gE16, B, 0, stream>>>(lin, src, dst, dinv, agg, N_EDGES);
        combine_relu_stats<<<(N_NODES + 3) / 4, B, 0, stream>>>(
            lin, agg, dinv, bl[l], act, stats, N_NODES);
        bn_apply<<<gNH, B, 0, stream>>>(act, stats, gl[l], bel[l], N_NODES);
    }

    // pooled output
    const long no = (long)NUM_GRAPHS * HID;
    zero_f32<<<(int)((no + B - 1) / B), B, 0, stream>>>(out, no);
    pool_scatter<<<gNH, B, 0, stream>>>(act, batch, out, N_NODES);
}